// S4DKernel_309237645920
// MI455X (gfx1250) — compile-verified
//
#include <hip/hip_runtime.h>

typedef __attribute__((ext_vector_type(2))) float v2f;
typedef __attribute__((ext_vector_type(4))) float v4f;
typedef __attribute__((ext_vector_type(8))) float v8f;

#define S4D_H   256
#define S4D_NH  32          // N/2
#define S4D_L   16384
#define S4D_T   128         // l_lo block (GEMM columns)
#define S4D_G   128         // l_hi count = L/T (GEMM rows)
#define S4D_LOG2E 1.44269504088896340736f

// ---------------------------------------------------------------------------
// Kernel 1: per-(h,n) coefficients, packed per n-pair as float4:
//   (dtA[2p]*log2e, dtA[2p+1]*log2e, 2*C_eff[2p], 2*C_eff[2p+1])
// 4096 entries (64 KB) in d_ws. Replaces per-wave recompute + IEEE division.
// ---------------------------------------------------------------------------
__global__ __launch_bounds__(256)
void s4d_precompute(const float* __restrict__ log_dt,
                    const float* __restrict__ C,
                    const float* __restrict__ log_A_real,
                    const float* __restrict__ A_imag,
                    v4f* __restrict__ tab) {
    const int t = blockIdx.x * blockDim.x + threadIdx.x;  // pair id 0..4095
    const int h = t >> 4;
    const int n = (t & 15) << 1;
    const int base = h * S4D_NH + n;

    const float dt  = __expf(log_dt[h]);
    const float ar0 = -__expf(log_A_real[base]);
    const float ar1 = -__expf(log_A_real[base + 1]);
    const float ai0 = A_imag[base];
    const float ai1 = A_imag[base + 1];
    const float dtA0 = ar0 * dt;
    const float dtA1 = ar1 * dt;
    const float ce0 = 2.0f * C[base]     * (__expf(dtA0) - 1.0f) * ar0 *
                      __builtin_amdgcn_rcpf(ar0 * ar0 + ai0 * ai0);
    const float ce1 = 2.0f * C[base + 1] * (__expf(dtA1) - 1.0f) * ar1 *
                      __builtin_amdgcn_rcpf(ar1 * ar1 + ai1 * ai1);

    v4f e;
    e.x = dtA0 * S4D_LOG2E;
    e.y = dtA1 * S4D_LOG2E;
    e.z = ce0;
    e.w = ce1;
    tab[t] = e;
}

// ---------------------------------------------------------------------------
// Kernel 2: per head h, K[h, i*T+j] = sum_n A[i,n]*B[n,j] with
//   A[i,n] = 2*C_eff[n] * 2^(d[n]*T*i),  B[n,j] = 2^(d[n]*j),  d = dtA*log2e
// One block per head (8 waves). Wave w owns j-tile w. B operands, the
// coefficient table, the it=0 A operands and the per-K geometric step
// factors 2^(d*T*16) are computed ONCE; the i-tile loop then advances A
// with a single v_pk_mul per K-step (geometric recurrence) + one WMMA.
// Trans-op count per wave: 48 (vs 144 with per-tile exps).
// ---------------------------------------------------------------------------
__global__ __launch_bounds__(256)
void s4d_wmma_main(const v4f* __restrict__ tab, float* __restrict__ out) {
    const int h    = blockIdx.x;
    const int wave = threadIdx.x >> 5;
    const int lane = threadIdx.x & 31;
    const int half = lane >> 4;          // 0: lanes 0-15, 1: lanes 16-31
    const int idx  = lane & 15;
    const int col  = (wave << 4) + idx;  // global j for this lane (B / D column)
    const float colf  = (float)col;
    const float rowf0 = (float)(S4D_T * idx);        // T * i at it = 0
    const float stepf = (float)(S4D_T * 16);         // T * 16 per i-tile

    // A/B lane layout for V_WMMA_F32_16X16X4_F32: K pair = (4k+2*half, +1),
    // M (resp. N) = idx. Pair index in table = 2k + half.
    v4f pk[8];
    v2f bv[8];   // B operands (it-invariant)
    v2f av[8];   // A operands for current i-tile (geometric recurrence)
    v2f st[8];   // per-K step factor 2^(d * T * 16)
#pragma unroll
    for (int k = 0; k < 8; ++k) {
        pk[k]   = tab[h * 16 + 2 * k + half];
        bv[k].x = __builtin_amdgcn_exp2f(pk[k].x * colf);
        bv[k].y = __builtin_amdgcn_exp2f(pk[k].y * colf);
        av[k].x = pk[k].z * __builtin_amdgcn_exp2f(pk[k].x * rowf0);
        av[k].y = pk[k].w * __builtin_amdgcn_exp2f(pk[k].y * rowf0);
        st[k].x = __builtin_amdgcn_exp2f(pk[k].x * stepf);
        st[k].y = __builtin_amdgcn_exp2f(pk[k].y * stepf);
    }

    float* outp = out + (size_t)h * S4D_L + col;
#pragma unroll
    for (int it = 0; it < 8; ++it) {
        v8f acc = {};
#pragma unroll
        for (int k = 0; k < 8; ++k) {
            // (neg_a, A, neg_b, B, c_mod, C, reuse_a, reuse_b)
            acc = __builtin_amdgcn_wmma_f32_16x16x4_f32(
                false, av[k], false, bv[k], (short)0, acc, false, false);
        }
        if (it < 7) {
#pragma unroll
            for (int k = 0; k < 8; ++k) av[k] = av[k] * st[k];  // v_pk_mul_f32
        }
        // D layout: VGPR v -> row (16*it + v + 8*half); row-major out, stride T
#pragma unroll
        for (int v = 0; v < 8; ++v) {
            outp[(size_t)(16 * it + v + 8 * half) * S4D_T] = acc[v];
        }
    }
}

extern "C" void kernel_launch(void* const* d_in, const int* in_sizes, int n_in,
                              void* d_out, int out_size, void* d_ws, size_t ws_size,
                              hipStream_t stream) {
    (void)in_sizes; (void)n_in; (void)out_size; (void)ws_size;
    const float* log_dt     = (const float*)d_in[0];
    const float* C          = (const float*)d_in[1];
    const float* log_A_real = (const float*)d_in[2];
    const float* A_imag     = (const float*)d_in[3];
    // d_in[4] is L (scalar) -- baked in as compile-time constant 16384.
    float* out = (float*)d_out;
    v4f*   tab = (v4f*)d_ws;   // 4096 * 16 B = 64 KB scratch

    // H*NH/2 = 4096 coefficient pairs
    s4d_precompute<<<dim3(16), dim3(256), 0, stream>>>(log_dt, C, log_A_real,
                                                       A_imag, tab);
    // one block per head; 8 waves = 8 j-tiles; each wave loops 8 i-tiles
    s4d_wmma_main<<<dim3(S4D_H), dim3(256), 0, stream>>>(tab, out);
}